// WAttention_19765439497051
// MI455X (gfx1250) — compile-verified
//
#include <hip/hip_runtime.h>
#include <math.h>

typedef __attribute__((ext_vector_type(2))) float v2f;
typedef __attribute__((ext_vector_type(8))) float v8f;

#define Qn 2048
#define Dn 512
#define Wn 5
#define Sn 64

__device__ __forceinline__ v8f wmma_f32(v2f a, v2f b, v8f c) {
  // D = A(16x4 f32) * B(4x16 f32) + C(16x16 f32)
  return __builtin_amdgcn_wmma_f32_16x16x4_f32(
      /*neg_a=*/false, a, /*neg_b=*/false, b,
      /*c_mod=*/(short)0, c, /*reuse_a=*/false, /*reuse_b=*/false);
}

// One wave computes Out[m0 : m0+16*MT, n0 : n0+16*NT] = X * Wm.
// X row-major (ldx), Wm row-major (ldw), Out row-major (ldo).
// Per K-step of 4: MT b64 A-loads + 2*NT b32 B-loads feed MT*NT WMMAs.
template <int MT, int NT>
__device__ __forceinline__ void gemm_tile(const float* __restrict__ X, int ldx,
                                          const float* __restrict__ Wm, int ldw,
                                          float* __restrict__ Out, int ldo,
                                          int m0, int n0, int K) {
  const int lane = threadIdx.x & 31;
  const int r16  = lane & 15;   // M (A,C/D) or N (B) within a 16x16 subtile
  const int hi   = lane >> 4;   // lane-half -> K pair (A,B) / row half (C)
  const int kb   = hi << 1;

  const float* xrow[MT];
  #pragma unroll
  for (int mi = 0; mi < MT; ++mi)
    xrow[mi] = X + (long)(m0 + 16 * mi + r16) * ldx;
  const float* wcol[NT];
  #pragma unroll
  for (int ni = 0; ni < NT; ++ni)
    wcol[ni] = Wm + n0 + 16 * ni + r16;

  // Warm the cache path for the streamed panels (global_prefetch_b8).
  __builtin_prefetch(xrow[0], 0, 1);
  __builtin_prefetch(wcol[0], 0, 1);

  v8f c[MT][NT];
  #pragma unroll
  for (int mi = 0; mi < MT; ++mi)
    #pragma unroll
    for (int ni = 0; ni < NT; ++ni)
      c[mi][ni] = (v8f){0.f, 0.f, 0.f, 0.f, 0.f, 0.f, 0.f, 0.f};

  #pragma unroll 2
  for (int k0 = 0; k0 < K; k0 += 4) {
    v2f a[MT];
    #pragma unroll
    for (int mi = 0; mi < MT; ++mi)
      a[mi] = *(const v2f*)(xrow[mi] + k0 + kb);
    v2f b[NT];
    #pragma unroll
    for (int ni = 0; ni < NT; ++ni) {
      b[ni].x = wcol[ni][(long)(k0 + kb)     * ldw];
      b[ni].y = wcol[ni][(long)(k0 + kb + 1) * ldw];
    }
    #pragma unroll
    for (int mi = 0; mi < MT; ++mi)
      #pragma unroll
      for (int ni = 0; ni < NT; ++ni)
        c[mi][ni] = wmma_f32(a[mi], b[ni], c[mi][ni]);
  }

  // C layout: VGPR r, lanes 0-15 -> M=r, lanes 16-31 -> M=r+8; N = lane&15
  #pragma unroll
  for (int mi = 0; mi < MT; ++mi)
    #pragma unroll
    for (int ni = 0; ni < NT; ++ni) {
      float* obase = Out + (long)(m0 + 16 * mi + 8 * hi) * ldo + n0 + 16 * ni + r16;
      #pragma unroll
      for (int r = 0; r < 8; ++r) obase[(long)r * ldo] = c[mi][ni][r];
    }
}

// Kernel 1: new_query = query @ Wq  (32x32 macro-tile per wave)
__global__ void __launch_bounds__(32) k_projQ(const float* __restrict__ query,
                                              const float* __restrict__ Wq,
                                              float* __restrict__ nq) {
  gemm_tile<2, 2>(query, Dn, Wq, Dn, nq, Dn, blockIdx.x * 32, blockIdx.y * 32, Dn);
}

// Kernel 2: new_key = key@Wk, new_value = key@Wv (z: 0 -> K, 1 -> V)
__global__ void __launch_bounds__(32) k_projKV(const float* __restrict__ key,
                                               const float* __restrict__ Wk,
                                               const float* __restrict__ Wv,
                                               float* __restrict__ nk,
                                               float* __restrict__ nv) {
  const int w  = blockIdx.y;
  const int mt = blockIdx.x >> 4;     // 0..1   (S/32)
  const int nt = blockIdx.x & 15;     // 0..15  (D/32)
  const float* X   = key + (long)w * Sn * Dn;
  const float* Wm  = (blockIdx.z ? Wv : Wk) + (long)w * Dn * Dn;
  float*       Out = (blockIdx.z ? nv : nk) + (long)w * Sn * Dn;
  gemm_tile<2, 2>(X, Dn, Wm, Dn, Out, Dn, mt * 32, nt * 32, Dn);
}

// Kernel 3: in-place L2-normalize rows of nq (2048 rows) then nk (320 rows).
// 256-thread block = 8 wave32; one wave per row of 512 floats.
__global__ void __launch_bounds__(256) k_l2rows(float* __restrict__ nq,
                                                float* __restrict__ nk) {
  const int wave = threadIdx.x >> 5;
  const int lane = threadIdx.x & 31;
  const long row = (long)blockIdx.x * 8 + wave;
  float* base;
  if (row < Qn)                base = nq + row * Dn;
  else if (row < Qn + Wn * Sn) base = nk + (row - Qn) * Dn;
  else return;

  float s = 0.f;
  #pragma unroll
  for (int j = 0; j < Dn / 32; ++j) { float v = base[lane + 32 * j]; s += v * v; }
  #pragma unroll
  for (int m = 16; m >= 1; m >>= 1) s += __shfl_xor(s, m, 32);
  const float inv = 1.0f / fmaxf(sqrtf(s), 1e-12f);
  #pragma unroll
  for (int j = 0; j < Dn / 32; ++j) base[lane + 32 * j] *= inv;
}

// Kernel 4: per (q-tile of 32 rows, w):
//   sim[32x64] = nq_tile @ nk_w^T   (WMMA; B fragment is contiguous since B = nk^T)
//   L2-normalize sim rows over S=64 (in LDS; one lane per row)
//   proto[32x512] = sim @ nv_w      (WMMA; A fragments from LDS, B strided loads
//                                    amortized over 2 M-subtiles)
__global__ void __launch_bounds__(32) k_simProto(const float* __restrict__ nq,
                                                 const float* __restrict__ nk,
                                                 const float* __restrict__ nv,
                                                 float* __restrict__ out) {
  __shared__ float sim[32][Sn + 4];   // row stride 68 floats (272B, 8B-aligned)

  const int lane = threadIdx.x & 31;
  const int r16  = lane & 15;
  const int hi   = lane >> 4;
  const int kb   = hi << 1;
  const int m0   = blockIdx.x * 32;
  const int w    = blockIdx.y;

  const float* nk_w = nk + (long)w * Sn * Dn;
  const float* nv_w = nv + (long)w * Sn * Dn;
  const float* qrow[2];
  #pragma unroll
  for (int mi = 0; mi < 2; ++mi)
    qrow[mi] = nq + (long)(m0 + 16 * mi + r16) * Dn;

  __builtin_prefetch(nk_w, 0, 1);
  __builtin_prefetch(nv_w, 0, 1);

  // ---- Phase 1: sim tiles (4 N-subtiles of 16, K = D = 512) ----
  for (int nt = 0; nt < Sn / 16; ++nt) {
    const float* krow = nk_w + (long)(nt * 16 + r16) * Dn;  // lane = N -> nk row
    v8f c[2];
    c[0] = (v8f){0.f, 0.f, 0.f, 0.f, 0.f, 0.f, 0.f, 0.f};
    c[1] = (v8f){0.f, 0.f, 0.f, 0.f, 0.f, 0.f, 0.f, 0.f};
    #pragma unroll 4
    for (int k0 = 0; k0 < Dn; k0 += 4) {
      v2f b = *(const v2f*)(krow + k0 + kb);                // shared across mi
      #pragma unroll
      for (int mi = 0; mi < 2; ++mi) {
        v2f a = *(const v2f*)(qrow[mi] + k0 + kb);
        c[mi] = wmma_f32(a, b, c[mi]);
      }
    }
    #pragma unroll
    for (int mi = 0; mi < 2; ++mi)
      #pragma unroll
      for (int r = 0; r < 8; ++r)
        sim[16 * mi + 8 * hi + r][nt * 16 + r16] = c[mi][r];
  }
  __syncthreads();

  // ---- Phase 2: normalize each of the 32 rows over S=64 (lane = row) ----
  {
    float s = 0.f;
    #pragma unroll
    for (int j = 0; j < Sn; ++j) { float v = sim[lane][j]; s += v * v; }
    const float inv = 1.0f / fmaxf(sqrtf(s), 1e-12f);
    #pragma unroll
    for (int j = 0; j < Sn; ++j) sim[lane][j] *= inv;
  }
  __syncthreads();

  // ---- Phase 3: proto = sim(32x64) @ nv_w(64x512) ----
  for (int nt = 0; nt < Dn / 16; ++nt) {
    const int n0 = nt * 16;
    v8f c[2];
    c[0] = (v8f){0.f, 0.f, 0.f, 0.f, 0.f, 0.f, 0.f, 0.f};
    c[1] = (v8f){0.f, 0.f, 0.f, 0.f, 0.f, 0.f, 0.f, 0.f};
    #pragma unroll
    for (int k0 = 0; k0 < Sn; k0 += 4) {
      v2f b;                                                 // shared across mi
      b.x = nv_w[(long)(k0 + kb)     * Dn + n0 + r16];
      b.y = nv_w[(long)(k0 + kb + 1) * Dn + n0 + r16];
      #pragma unroll
      for (int mi = 0; mi < 2; ++mi) {
        v2f a = *(const v2f*)(&sim[16 * mi + r16][k0 + kb]); // 8B-aligned LDS read
        c[mi] = wmma_f32(a, b, c[mi]);
      }
    }
    // out layout [B=1, Q, W, D]
    #pragma unroll
    for (int mi = 0; mi < 2; ++mi)
      #pragma unroll
      for (int r = 0; r < 8; ++r) {
        const int M = 16 * mi + 8 * hi + r;
        out[((long)(m0 + M) * Wn + w) * Dn + n0 + r16] = c[mi][r];
      }
  }
}

extern "C" void kernel_launch(void* const* d_in, const int* in_sizes, int n_in,
                              void* d_out, int out_size, void* d_ws, size_t ws_size,
                              hipStream_t stream) {
  const float* query = (const float*)d_in[0]; // [1,2048,512]
  const float* key   = (const float*)d_in[1]; // [1,5,64,512]
  const float* Wq    = (const float*)d_in[2]; // [512,512]
  const float* Wk    = (const float*)d_in[3]; // [5,512,512]
  const float* Wv    = (const float*)d_in[4]; // [5,512,512]
  float* out = (float*)d_out;                 // [1,2048,5,512]

  float* ws = (float*)d_ws;
  float* nq = ws;                              // 2048*512
  float* nk = nq + (size_t)Qn * Dn;            // 5*64*512
  float* nv = nk + (size_t)Wn * Sn * Dn;       // 5*64*512

  const dim3 wave(32);
  k_projQ  <<<dim3(Qn / 32, Dn / 32),             wave,      0, stream>>>(query, Wq, nq);
  k_projKV <<<dim3((Sn / 32) * (Dn / 32), Wn, 2), wave,      0, stream>>>(key, Wk, Wv, nk, nv);
  const int rows = Qn + Wn * Sn;               // 2368 rows, 8 per block
  k_l2rows <<<dim3((rows + 7) / 8),               dim3(256), 0, stream>>>(nq, nk);
  k_simProto<<<dim3(Qn / 32, Wn),                 wave,      0, stream>>>(nq, nk, nv, out);
}